// GraphPoolingHierarchy_30631706755212
// MI455X (gfx1250) — compile-verified
//
#include <hip/hip_runtime.h>
#include <hip/hip_bf16.h>

typedef float v2f __attribute__((ext_vector_type(2)));
typedef float v8f __attribute__((ext_vector_type(8)));

#define DHID 128
#define BGR  128
#define NPGC 4096
#define K1   2868
#define K2   1434
#define K3   431

// ---------------------------------------------------------------------------
// Passthrough copy of x into the first output segment (float4 vectorized).
// ---------------------------------------------------------------------------
__global__ __launch_bounds__(256) void copy_f4_kernel(const float4* __restrict__ src,
                                                      float4* __restrict__ dst,
                                                      size_t n4) {
    size_t i = (size_t)blockIdx.x * blockDim.x + threadIdx.x;
    size_t stride = (size_t)gridDim.x * blockDim.x;
    for (; i < n4; i += stride) dst[i] = src[i];
}

// ---------------------------------------------------------------------------
// Per-graph fused score + bitonic top-k. One workgroup per graph.
// key = (~ordered(score) << 32) | index  -> ascending sort == descending score,
// ties broken by ascending index (matches jax.lax.top_k).
// Emits kept-node index and tanh(score * rsqrt(w.w)) per kept row.
// ---------------------------------------------------------------------------
__device__ __forceinline__ unsigned ord_f32(float f) {
    unsigned u = __float_as_uint(f);
    return (u & 0x80000000u) ? ~u : (u | 0x80000000u);
}

__global__ __launch_bounds__(512) void topk_sortscore_kernel(
    const float* __restrict__ cur,   // [BGR, n, DHID]
    const float* __restrict__ w,     // [DHID]
    int n, int npow2, int k,
    int* __restrict__ idxOut,        // [BGR*k]
    float* __restrict__ scaleOut)    // [BGR*k]
{
    __shared__ unsigned long long keys[4096];
    __shared__ float wsh[DHID];
    __shared__ float s_inv;

    const int tid = threadIdx.x;
    const int b = blockIdx.x;

    if (tid < DHID) wsh[tid] = w[tid];
    __syncthreads();
    if (tid == 0) {
        float s = 0.f;
        for (int i = 0; i < DHID; ++i) s += wsh[i] * wsh[i];
        s_inv = rsqrtf(s);
    }

    const float* base = cur + (size_t)b * n * DHID;
    for (int i = tid; i < npow2; i += 512) {
        unsigned long long key = 0xFFFFFFFFFFFFFFFFull;  // padding sorts last
        if (i < n) {
            const float4* row = (const float4*)(base + (size_t)i * DHID);
            float s = 0.f;
#pragma unroll 8
            for (int j = 0; j < DHID / 4; ++j) {
                float4 v = row[j];
                s += v.x * wsh[4 * j] + v.y * wsh[4 * j + 1] +
                     v.z * wsh[4 * j + 2] + v.w * wsh[4 * j + 3];
            }
            key = ((unsigned long long)(~ord_f32(s)) << 32) | (unsigned)i;
        }
        keys[i] = key;
    }
    __syncthreads();

    // Bitonic sort (ascending) over npow2 64-bit keys in LDS.
    for (int size = 2; size <= npow2; size <<= 1) {
        for (int stride = size >> 1; stride > 0; stride >>= 1) {
            for (int i = tid; i < npow2; i += 512) {
                int ixj = i ^ stride;
                if (ixj > i) {
                    bool asc = ((i & size) == 0);
                    unsigned long long a = keys[i], c = keys[ixj];
                    if ((a > c) == asc) { keys[i] = c; keys[ixj] = a; }
                }
            }
            __syncthreads();
        }
    }

    // Emit top-k: decode index and score, apply tanh(score/||w||).
    for (int j = tid; j < k; j += 512) {
        unsigned long long key = keys[j];
        unsigned idx = (unsigned)(key & 0xFFFFFFFFu);
        unsigned o = ~(unsigned)(key >> 32);
        unsigned bits = (o & 0x80000000u) ? (o & 0x7FFFFFFFu) : ~o;
        float s = __uint_as_float(bits);
        idxOut[b * k + j] = (int)idx;
        scaleOut[b * k + j] = tanhf(s * s_inv);
    }
}

// ---------------------------------------------------------------------------
// out[r, :] = (cur[g, idx[r], :] * scale[r]) @ W^T + bias
// using V_WMMA_F32_16X16X4_F32. One wave per 16x16 output tile; 8 waves per
// block cover the full N=128; gather+tanh scaling fused into A-fragment load.
// M = BGR*k is always a multiple of 16 (BGR=128), N=K=128 -> no tails, EXEC
// stays all-ones as WMMA requires.
//
// Fragment layouts (ISA 7.12.2, wave32):
//   A 16x4 f32 : lane L holds A[L&15][2*(L>>4) + v], v in {0,1}
//   B 4x16 f32 : lane L holds B[2*(L>>4) + v][L&15]
//   C/D 16x16  : lane L, reg r holds D[r + 8*(L>>4)][L&15]
// ---------------------------------------------------------------------------
__global__ __launch_bounds__(256) void pool_gemm_wmma_kernel(
    const float* __restrict__ cur,       // [BGR, n, DHID] gather source
    const int* __restrict__ idxArr,      // [BGR*k]
    const float* __restrict__ scaleArr,  // [BGR*k]
    const float* __restrict__ W,         // [DHID, DHID] row-major
    const float* __restrict__ bias,      // [DHID]
    float* __restrict__ outSeg,          // [BGR*k, DHID]
    int n, int k)
{
    const int lane = threadIdx.x & 31;
    const int wave = threadIdx.x >> 5;
    const int m = lane & 15;   // A row / B col / D col within tile
    const int h = lane >> 4;   // selects K pair within fragment

    const int rowBase = blockIdx.x * 16;
    const int r = rowBase + m;
    const int g = r / k;                       // graph id
    const int local = idxArr[r];               // kept node within graph
    const float scale = scaleArr[r];
    const float* __restrict__ arow = cur + ((size_t)g * n + local) * DHID;

    const int n0 = wave * 16;
    const float* __restrict__ brow = W + (size_t)(n0 + m) * DHID;  // B[k][n]=W[n0+n][k]

    v8f acc = {};
#pragma unroll
    for (int kb = 0; kb < DHID; kb += 4) {
        v2f a, bf;
        a.x = arow[kb + 2 * h] * scale;
        a.y = arow[kb + 2 * h + 1] * scale;
        bf.x = brow[kb + 2 * h];
        bf.y = brow[kb + 2 * h + 1];
        acc = __builtin_amdgcn_wmma_f32_16x16x4_f32(
            /*neg_a=*/false, a, /*neg_b=*/false, bf,
            /*c_mod=*/(short)0, acc, /*reuse_a=*/false, /*reuse_b=*/false);
    }

    const float bv = bias[n0 + m];
#pragma unroll
    for (int rr = 0; rr < 8; ++rr) {
        int orow = rowBase + rr + 8 * h;
        outSeg[(size_t)orow * DHID + (n0 + m)] = acc[rr] + bv;
    }
}

// ---------------------------------------------------------------------------
extern "C" void kernel_launch(void* const* d_in, const int* in_sizes, int n_in,
                              void* d_out, int out_size, void* d_ws, size_t ws_size,
                              hipStream_t stream) {
    (void)in_sizes; (void)n_in; (void)out_size; (void)ws_size;

    const float* x  = (const float*)d_in[0];
    // d_in[1] edge_index, d_in[2] batch: never affect outputs (see reference)
    const float* w1 = (const float*)d_in[3];
    const float* W1 = (const float*)d_in[4];
    const float* b1 = (const float*)d_in[5];
    const float* w2 = (const float*)d_in[6];
    const float* W2 = (const float*)d_in[7];
    const float* b2 = (const float*)d_in[8];
    const float* w3 = (const float*)d_in[9];
    const float* W3 = (const float*)d_in[10];
    const float* b3 = (const float*)d_in[11];

    float* out = (float*)d_out;
    const size_t off1 = (size_t)BGR * NPGC * DHID;          // after x
    const size_t off2 = off1 + (size_t)BGR * K1 * DHID;     // after level-1
    const size_t off3 = off2 + (size_t)BGR * K2 * DHID;     // after level-2

    // workspace: idx + scale arrays, sized for largest level (K1)
    int*   idxArr   = (int*)d_ws;
    float* scaleArr = (float*)((int*)d_ws + (size_t)BGR * K1);

    // feats[0] = x (passthrough copy)
    copy_f4_kernel<<<8192, 256, 0, stream>>>(
        (const float4*)x, (float4*)out, (size_t)BGR * NPGC * DHID / 4);

    // ---- Level 1: cur = x, n = 4096, k = 2868 ----
    topk_sortscore_kernel<<<BGR, 512, 0, stream>>>(x, w1, NPGC, 4096, K1,
                                                   idxArr, scaleArr);
    pool_gemm_wmma_kernel<<<(BGR * K1) / 16, 256, 0, stream>>>(
        x, idxArr, scaleArr, W1, b1, out + off1, NPGC, K1);

    // ---- Level 2: cur = level-1 output, n = 2868, k = 1434 ----
    topk_sortscore_kernel<<<BGR, 512, 0, stream>>>(out + off1, w2, K1, 4096, K2,
                                                   idxArr, scaleArr);
    pool_gemm_wmma_kernel<<<(BGR * K2) / 16, 256, 0, stream>>>(
        out + off1, idxArr, scaleArr, W2, b2, out + off2, K1, K2);

    // ---- Level 3: cur = level-2 output, n = 1434, k = 431 ----
    topk_sortscore_kernel<<<BGR, 512, 0, stream>>>(out + off2, w3, K2, 2048, K3,
                                                   idxArr, scaleArr);
    pool_gemm_wmma_kernel<<<(BGR * K3) / 16, 256, 0, stream>>>(
        out + off2, idxArr, scaleArr, W3, b3, out + off3, K2, K3);
}